// HashEncoding_4552665334153
// MI455X (gfx1250) — compile-verified
//
#include <hip/hip_runtime.h>
#include <cmath>

// native clang vector types (required by __builtin_nontemporal_* and for clean b64/b128 VMEM)
typedef float v2f __attribute__((ext_vector_type(2)));
typedef float v4f __attribute__((ext_vector_type(4)));

// ---------------- compile-time level metadata (matches reference _level_metas) ---
static constexpr int NLVL = 16;

// size = min(ceil(res^3/8)*8, 2^19); levels 5..15 are hashed (size == 2^19)
static constexpr unsigned kSize[NLVL] = {
    4096u,   12168u,  29792u,  79512u,  205384u,
    524288u, 524288u, 524288u, 524288u, 524288u, 524288u,
    524288u, 524288u, 524288u, 524288u, 524288u};

// float-element offset of each level's table; level table is [8*size, 2] floats
// (prefix sums of 16*size; total = 97,569,920 == TOTAL_PARAMS)
static constexpr unsigned kOff[NLVL] = {
    0u,        65536u,    260224u,   736896u,   2009088u,
    5295232u,  13683840u, 22072448u, 30461056u, 38849664u,
    47238272u, 55626880u, 64015488u, 72404096u, 80792704u, 89181312u};

// resolution (only used by dense levels 0..4)
static constexpr int kRes[NLVL] = {16, 23, 31, 43, 59,
                                   81, 112, 154, 213, 294, 407,
                                   562, 777, 1073, 1483, 2049};

static constexpr unsigned H1 = 2654435761u;  // y hash constant
static constexpr unsigned H2 = 805459861u;   // z hash constant

struct ScaleArgs { float s[NLVL]; };

// live gather state for one level (becomes registers after full unroll)
struct LevelData {
    v2f   t[8];
    float dx, dy, dz;
};

// ---------------- per-level gather: issue 8 independent b64 loads -----------------
template <int L>
__device__ __forceinline__ void level_gather(const float* __restrict__ params,
                                             float x, float y, float z, float scale,
                                             LevelData& d) {
    constexpr unsigned size = kSize[L];
    constexpr unsigned off  = kOff[L];
    constexpr bool hashed   = (size >= (1u << 19));

    const float px = fmaf(x, scale, 0.5f);
    const float py = fmaf(y, scale, 0.5f);
    const float pz = fmaf(z, scale, 0.5f);
    const float gx = floorf(px), gy = floorf(py), gz = floorf(pz);
    d.dx = px - gx; d.dy = py - gy; d.dz = pz - gz;
    const unsigned ix = (unsigned)(int)gx;   // inputs in [0,1) -> coords >= 0
    const unsigned iy = (unsigned)(int)gy;
    const unsigned iz = (unsigned)(int)gz;

    // subgrid parity of corner c is just base parity XOR c
    const unsigned sub0 = ((ix & 1u) << 2) | ((iy & 1u) << 1) | (iz & 1u);

    // separable index terms: 2 muls per level instead of 16
    unsigned y0, y1, z0, z1;
    if constexpr (hashed) {
        y0 = iy * H1;  y1 = y0 + H1;
        z0 = iz * H2;  z1 = z0 + H2;
    } else {
        constexpr unsigned r = (unsigned)kRes[L];
        y0 = iy * r;       y1 = y0 + r;
        z0 = iz * (r * r); z1 = z0 + r * r;
    }

    const float* __restrict__ tab = params + off;
#pragma unroll
    for (int c = 0; c < 8; ++c) {
        const unsigned bx = ((unsigned)c >> 2) & 1u;
        const unsigned by = ((unsigned)c >> 1) & 1u;
        const unsigned bz = (unsigned)c & 1u;
        unsigned idx;
        if constexpr (hashed) {
            idx = ((ix + bx) ^ (by ? y1 : y0) ^ (bz ? z1 : z0)) & (size - 1u);
        } else {
            idx = ((ix + bx) + (by ? y1 : y0) + (bz ? z1 : z0)) % size;
        }
        const unsigned flat = (sub0 ^ (unsigned)c) * size + idx;
        d.t[c] = *reinterpret_cast<const v2f*>(tab + 2ull * flat);
    }
}

// ---------------- trilinear reduce of one level's 8 corners -----------------------
__device__ __forceinline__ v2f level_reduce(const LevelData& d) {
    const float wx[2] = {1.0f - d.dx, d.dx};
    const float wy[2] = {1.0f - d.dy, d.dy};
    const float wz[2] = {1.0f - d.dz, d.dz};
    float a0 = 0.0f, a1 = 0.0f;
#pragma unroll
    for (int c = 0; c < 8; ++c) {
        const float w = wx[(c >> 2) & 1] * wy[(c >> 1) & 1] * wz[c & 1];
        a0 = fmaf(w, d.t[c].x, a0);
        a1 = fmaf(w, d.t[c].y, a1);
    }
    v2f r; r.x = a0; r.y = a1;
    return r;
}

// ---------------- 4 levels in flight: 32 outstanding b64 gathers per wave ---------
template <int L>
__device__ __forceinline__ void do_quad(const float* __restrict__ params,
                                        float* __restrict__ out, unsigned p,
                                        float x, float y, float z,
                                        const ScaleArgs& sc) {
    LevelData d0, d1, d2, d3;
    level_gather<L + 0>(params, x, y, z, sc.s[L + 0], d0);
    level_gather<L + 1>(params, x, y, z, sc.s[L + 1], d1);
    level_gather<L + 2>(params, x, y, z, sc.s[L + 2], d2);
    level_gather<L + 3>(params, x, y, z, sc.s[L + 3], d3);

    const v2f r0 = level_reduce(d0);
    const v2f r1 = level_reduce(d1);
    const v2f r2 = level_reduce(d2);
    const v2f r3 = level_reduce(d3);

    v4f lo; lo.x = r0.x; lo.y = r0.y; lo.z = r1.x; lo.w = r1.y;
    v4f hi; hi.x = r2.x; hi.y = r2.y; hi.z = r3.x; hi.w = r3.y;

    // streaming output: non-temporal (TH=NT) so 128MB of writes don't evict
    // the hash tables from the 192MB L2
    float* base = out + (size_t)p * 32u + (size_t)(2 * L);
    __builtin_nontemporal_store(lo, reinterpret_cast<v4f*>(base));
    __builtin_nontemporal_store(hi, reinterpret_cast<v4f*>(base + 4));
}

__global__ __launch_bounds__(256) void hashgrid_encode_kernel(
    const float* __restrict__ inp, const float* __restrict__ params,
    float* __restrict__ out, unsigned n, ScaleArgs sc) {
    const unsigned p = blockIdx.x * 256u + threadIdx.x;
    if (p >= n) return;
    // inputs are read exactly once -> non-temporal loads (merged into b96 NT)
    const float x = __builtin_nontemporal_load(inp + (size_t)p * 3u + 0u);
    const float y = __builtin_nontemporal_load(inp + (size_t)p * 3u + 1u);
    const float z = __builtin_nontemporal_load(inp + (size_t)p * 3u + 2u);

    do_quad<0>(params, out, p, x, y, z, sc);
    do_quad<4>(params, out, p, x, y, z, sc);
    do_quad<8>(params, out, p, x, y, z, sc);
    do_quad<12>(params, out, p, x, y, z, sc);
}

// ---------------- host launcher ---------------------------------------------------
extern "C" void kernel_launch(void* const* d_in, const int* in_sizes, int n_in,
                              void* d_out, int out_size, void* d_ws, size_t ws_size,
                              hipStream_t stream) {
    (void)n_in; (void)out_size; (void)d_ws; (void)ws_size;
    const float* inp    = (const float*)d_in[0];   // [N,3] f32
    const float* params = (const float*)d_in[1];   // [97,569,920] f32
    float* out          = (float*)d_out;           // [N,32] f32

    const unsigned n = (unsigned)(in_sizes[0] / 3);

    // scales computed exactly as the reference: 2^(i*log2(s)) * 16 - 1 (double -> f32)
    ScaleArgs sc;
    for (int i = 0; i < NLVL; ++i) {
        sc.s[i] = (float)(std::exp2((double)i * std::log2(1.38191288)) * 16.0 - 1.0);
    }

    dim3 block(256);
    dim3 grid((n + 255u) / 256u);
    hipLaunchKernelGGL(hashgrid_encode_kernel, grid, block, 0, stream,
                       inp, params, out, n, sc);
}